// EdgeUpdateBlock_6253472383849
// MI455X (gfx1250) — compile-verified
//
#include <hip/hip_runtime.h>
#include <hip/hip_bf16.h>

#define HIDDEN 128
#define HEADS 8
#define HEAD_DIM 16

typedef __attribute__((ext_vector_type(2))) float v2f;
typedef __attribute__((ext_vector_type(8))) float v8f;

// ---------------------------------------------------------------------------
// Y[r, 0:128] = X[r, 0:128] @ W[128,128] (+ bias). nrows must be a multiple of
// 16 (50000 and 500000 both are). One wave per 16-row slab, 4 waves per block.
// Uses V_WMMA_F32_16X16X4_F32: D[16x16] += A[16x4] @ B[4x16], fp32 end to end.
// ---------------------------------------------------------------------------
__global__ void gemm128_wmma_kernel(const float* __restrict__ X,
                                    const float* __restrict__ W,
                                    const float* __restrict__ bias,
                                    float* __restrict__ Y,
                                    int nrows) {
    const int wave  = threadIdx.x >> 5;
    const int lane  = threadIdx.x & 31;
    const int row0  = (blockIdx.x * 4 + wave) * 16;
    if (row0 >= nrows) return;

    const int lhalf = lane >> 4;   // 0: lanes 0-15, 1: lanes 16-31
    const int l     = lane & 15;

    // A-fragment source row for this lane (A 16x4: lanes 0-15 hold K=0,1;
    // lanes 16-31 hold K=2,3; M = lane%16).
    const float* __restrict__ xrow = X + (size_t)(row0 + l) * HIDDEN;

    v8f acc[8];
    const v8f vzero = {0.f, 0.f, 0.f, 0.f, 0.f, 0.f, 0.f, 0.f};
#pragma unroll
    for (int n = 0; n < 8; ++n) acc[n] = vzero;

    for (int k0 = 0; k0 < HIDDEN; k0 += 4) {
        v2f a;
        a.x = xrow[k0 + 2 * lhalf + 0];
        a.y = xrow[k0 + 2 * lhalf + 1];
        const float* __restrict__ wr0 = W + (size_t)(k0 + 2 * lhalf)     * HIDDEN;
        const float* __restrict__ wr1 = W + (size_t)(k0 + 2 * lhalf + 1) * HIDDEN;
#pragma unroll
        for (int n = 0; n < 8; ++n) {
            const int col = n * 16 + l;       // B 4x16: lanes hold N=col
            v2f b;
            b.x = wr0[col];
            b.y = wr1[col];
            acc[n] = __builtin_amdgcn_wmma_f32_16x16x4_f32(
                /*neg_a=*/false, a, /*neg_b=*/false, b,
                /*c_mod=*/(short)0, acc[n],
                /*reuse_a=*/false, /*reuse_b=*/false);
        }
    }

    // C/D 16x16 layout: VGPR j -> row row0+j (lanes 0-15) / row0+j+8 (16-31).
#pragma unroll
    for (int n = 0; n < 8; ++n) {
        const int col = n * 16 + l;
        const float bv = bias ? bias[col] : 0.0f;
#pragma unroll
        for (int j = 0; j < 8; ++j) {
            const int row = row0 + j + 8 * lhalf;
            Y[(size_t)row * HIDDEN + col] = acc[n][j] + bv;
        }
    }
}

// ---------------------------------------------------------------------------
// Init per-node softmax state: max=-inf, z=0, weighted-sum accumulator=0.
// ---------------------------------------------------------------------------
__global__ void init_state_kernel(float* __restrict__ nmax,
                                  float* __restrict__ z,
                                  float* __restrict__ acc,
                                  int nnodes) {
    const int i = blockIdx.x * blockDim.x + threadIdx.x;
    if (i < nnodes * HEADS) { nmax[i] = -INFINITY; z[i] = 0.0f; }
    if (i < nnodes * HIDDEN) acc[i] = 0.0f;
}

__device__ __forceinline__ void atomicMaxFloat(float* addr, float val) {
    unsigned int* ua  = (unsigned int*)addr;
    unsigned int  old = *ua;
    while (__uint_as_float(old) < val) {
        const unsigned int assumed = old;
        old = atomicCAS(ua, assumed, __float_as_uint(val));
        if (old == assumed) break;
    }
}

// ---------------------------------------------------------------------------
// Per (edge, head): alpha = <q[dst,h,:], k[src,h,:]+e[edge,h,:]> / sqrt(16),
// then segment-max into nmax[dst,h].
// ---------------------------------------------------------------------------
__global__ void alpha_kernel(const float* __restrict__ q,
                             const float* __restrict__ k,
                             const float* __restrict__ e,
                             const int* __restrict__ src,
                             const int* __restrict__ dst,
                             float* __restrict__ alpha,
                             float* __restrict__ nmax,
                             int nedges) {
    const long long idx = (long long)blockIdx.x * blockDim.x + threadIdx.x;
    if (idx >= (long long)nedges * HEADS) return;
    const int eidx = (int)(idx >> 3);
    const int h    = (int)(idx & 7);
    const int s    = src[eidx];
    const int d    = dst[eidx];
    const float* __restrict__ qp = q + (size_t)d * HIDDEN + h * HEAD_DIM;
    const float* __restrict__ kp = k + (size_t)s * HIDDEN + h * HEAD_DIM;
    const float* __restrict__ ep = e + (size_t)eidx * HIDDEN + h * HEAD_DIM;
    float a = 0.0f;
#pragma unroll
    for (int t = 0; t < HEAD_DIM; ++t) a += qp[t] * (kp[t] + ep[t]);
    a *= 0.25f;   // 1/sqrt(HEAD_DIM)
    alpha[idx] = a;
    atomicMaxFloat(&nmax[(size_t)d * HEADS + h], a);
}

// ---------------------------------------------------------------------------
// Per (edge, head): a = exp(alpha - m[dst,h]); z[dst,h] += a;
// acc[dst,h,:] += a * (v[src,h,:] + e[edge,h,:]).  Normalization by z deferred.
// ---------------------------------------------------------------------------
__global__ void scatter_kernel(const float* __restrict__ v,
                               const float* __restrict__ e,
                               const float* __restrict__ alpha,
                               const float* __restrict__ nmax,
                               const int* __restrict__ src,
                               const int* __restrict__ dst,
                               float* __restrict__ z,
                               float* __restrict__ acc,
                               int nedges) {
    const long long idx = (long long)blockIdx.x * blockDim.x + threadIdx.x;
    if (idx >= (long long)nedges * HEADS) return;
    const int eidx = (int)(idx >> 3);
    const int h    = (int)(idx & 7);
    const int s    = src[eidx];
    const int d    = dst[eidx];
    const float a = __expf(alpha[idx] - nmax[(size_t)d * HEADS + h]);
    atomicAdd(&z[(size_t)d * HEADS + h], a);
    const float* __restrict__ vp = v + (size_t)s * HIDDEN + h * HEAD_DIM;
    const float* __restrict__ ep = e + (size_t)eidx * HIDDEN + h * HEAD_DIM;
    float* __restrict__ ap = acc + (size_t)d * HIDDEN + h * HEAD_DIM;
#pragma unroll
    for (int t = 0; t < HEAD_DIM; ++t) atomicAdd(&ap[t], a * (vp[t] + ep[t]));
}

// ---------------------------------------------------------------------------
// One wave32 per node: normalize, beta-gated skip, LayerNorm, ReLU, residual.
// Lane owns columns {lane, lane+32, lane+64, lane+96}; reductions via shfl_xor.
// ---------------------------------------------------------------------------
__global__ void finalize_kernel(const float* __restrict__ edge_state,
                                const float* __restrict__ acc,
                                const float* __restrict__ z,
                                const float* __restrict__ xr,
                                const float* __restrict__ Wbeta,
                                const float* __restrict__ ln_gamma,
                                const float* __restrict__ ln_beta,
                                float* __restrict__ out,
                                int nnodes) {
    const int wave = threadIdx.x >> 5;
    const int lane = threadIdx.x & 31;
    const int node = blockIdx.x * (blockDim.x >> 5) + wave;
    if (node >= nnodes) return;

    float o[4], x[4];
    float bdot = 0.0f;
#pragma unroll
    for (int t = 0; t < 4; ++t) {
        const int c  = lane + 32 * t;
        const float zz = z[(size_t)node * HEADS + (c >> 4)];
        float ov = acc[(size_t)node * HIDDEN + c];
        ov = (zz > 0.0f) ? (ov / zz) : 0.0f;     // empty segments -> 0
        const float xv = xr[(size_t)node * HIDDEN + c];
        o[t] = ov; x[t] = xv;
        bdot += ov * Wbeta[c] + xv * Wbeta[HIDDEN + c] + (ov - xv) * Wbeta[2 * HIDDEN + c];
    }
#pragma unroll
    for (int off = 16; off > 0; off >>= 1) bdot += __shfl_xor(bdot, off, 32);
    const float beta = 1.0f / (1.0f + __expf(-bdot));

    float g[4], mu = 0.0f;
#pragma unroll
    for (int t = 0; t < 4; ++t) { g[t] = beta * x[t] + (1.0f - beta) * o[t]; mu += g[t]; }
#pragma unroll
    for (int off = 16; off > 0; off >>= 1) mu += __shfl_xor(mu, off, 32);
    mu *= (1.0f / HIDDEN);

    float var = 0.0f;
#pragma unroll
    for (int t = 0; t < 4; ++t) { const float dlt = g[t] - mu; var += dlt * dlt; }
#pragma unroll
    for (int off = 16; off > 0; off >>= 1) var += __shfl_xor(var, off, 32);
    var *= (1.0f / HIDDEN);
    const float rstd = rsqrtf(var + 1e-5f);

#pragma unroll
    for (int t = 0; t < 4; ++t) {
        const int c = lane + 32 * t;
        const float y = (g[t] - mu) * rstd * ln_gamma[c] + ln_beta[c];
        out[(size_t)node * HIDDEN + c] =
            edge_state[(size_t)node * HIDDEN + c] + fmaxf(y, 0.0f);
    }
}

// ---------------------------------------------------------------------------
extern "C" void kernel_launch(void* const* d_in, const int* in_sizes, int n_in,
                              void* d_out, int out_size, void* d_ws, size_t ws_size,
                              hipStream_t stream) {
    const float* edge_state = (const float*)d_in[0];
    const int*   lg_idx     = (const int*)d_in[1];
    const float* angle_emb  = (const float*)d_in[2];
    const float* Wq    = (const float*)d_in[3];
    const float* bq    = (const float*)d_in[4];
    const float* Wk    = (const float*)d_in[5];
    const float* bk    = (const float*)d_in[6];
    const float* Wv    = (const float*)d_in[7];
    const float* bv    = (const float*)d_in[8];
    const float* We    = (const float*)d_in[9];
    const float* Wskip = (const float*)d_in[10];
    const float* bskip = (const float*)d_in[11];
    const float* Wbeta = (const float*)d_in[12];
    const float* lng   = (const float*)d_in[13];
    const float* lnb   = (const float*)d_in[14];

    const int N = in_sizes[0] / HIDDEN;      // 50000
    const int E = in_sizes[2] / HIDDEN;      // 500000
    const int* src = lg_idx;                 // lg_edge_index[0]
    const int* dst = lg_idx + E;             // lg_edge_index[1]

    // Workspace layout (fp32), ~403 MB total.
    float* ws    = (float*)d_ws;
    float* q     = ws;
    float* k     = q     + (size_t)N * HIDDEN;
    float* v     = k     + (size_t)N * HIDDEN;
    float* xr    = v     + (size_t)N * HIDDEN;
    float* accum = xr    + (size_t)N * HIDDEN;
    float* e     = accum + (size_t)N * HIDDEN;
    float* alpha = e     + (size_t)E * HIDDEN;
    float* nmax  = alpha + (size_t)E * HEADS;
    float* z     = nmax  + (size_t)N * HEADS;

    // 1) Node GEMMs (q, k, v, skip) + edge GEMM (e) on the WMMA path.
    const int gridN = (N / 16 + 3) / 4;
    const int gridE = (E / 16 + 3) / 4;
    gemm128_wmma_kernel<<<gridN, 128, 0, stream>>>(edge_state, Wq,    bq,      q,  N);
    gemm128_wmma_kernel<<<gridN, 128, 0, stream>>>(edge_state, Wk,    bk,      k,  N);
    gemm128_wmma_kernel<<<gridN, 128, 0, stream>>>(edge_state, Wv,    bv,      v,  N);
    gemm128_wmma_kernel<<<gridN, 128, 0, stream>>>(edge_state, Wskip, bskip,   xr, N);
    gemm128_wmma_kernel<<<gridE, 128, 0, stream>>>(angle_emb,  We,    nullptr, e,  E);

    // 2) Init softmax state.
    init_state_kernel<<<(N * HIDDEN + 255) / 256, 256, 0, stream>>>(nmax, z, accum, N);

    // 3) Edge-parallel attention logits + segment max.
    const long long eh = (long long)E * HEADS;
    const int gridEH = (int)((eh + 255) / 256);
    alpha_kernel<<<gridEH, 256, 0, stream>>>(q, k, e, src, dst, alpha, nmax, E);

    // 4) Edge-parallel exp/scatter (unnormalized weighted sum + partition fn).
    scatter_kernel<<<gridEH, 256, 0, stream>>>(v, e, alpha, nmax, src, dst, z, accum, E);

    // 5) Node-parallel normalize + beta gate + LayerNorm + ReLU + residual.
    finalize_kernel<<<(N + 7) / 8, 256, 0, stream>>>(edge_state, accum, z, xr,
                                                     Wbeta, lng, lnb,
                                                     (float*)d_out, N);
}